// DifferentiableLogicLayer_27685359190337
// MI455X (gfx1250) — compile-verified
//
#include <hip/hip_runtime.h>
#include <math.h>

typedef __attribute__((ext_vector_type(2))) float v2f;
typedef __attribute__((ext_vector_type(4))) float f32x4;
typedef __attribute__((ext_vector_type(8))) float v8f;

#define PENALTY_LAMBDA 0.5f

// ---------------------------------------------------------------------------
// Kernel 1: gates = sigmoid(gate_logits); zero firing[]; zero pen_1d[]
// ---------------------------------------------------------------------------
__global__ void dll_init_kernel(const float* __restrict__ gate_logits,
                                float* __restrict__ gates,
                                float* __restrict__ firing,
                                float* __restrict__ pen,
                                int R, int V) {
    int t = blockIdx.x * blockDim.x + threadIdx.x;
    if (t < R) {
        float g = gate_logits[t];
        gates[t]  = 1.0f / (1.0f + __expf(-g));
        firing[t] = 0.0f;
    } else if (t < R + V) {
        pen[t - R] = 0.0f;
    }
}

// ---------------------------------------------------------------------------
// Kernel 2: scatter-add of gated penalties into pen_1d; mark firing rules.
// pen_1d (512 KB) is L2-resident -> GLOBAL_ATOMIC_ADD_F32 at L2 rate.
// ---------------------------------------------------------------------------
__global__ void dll_scatter_kernel(const int* __restrict__ rule_ids,
                                   const int* __restrict__ token_ids,
                                   const float* __restrict__ gates,
                                   float* __restrict__ firing,
                                   float* __restrict__ pen,
                                   int E) {
    int e = blockIdx.x * blockDim.x + threadIdx.x;
    if (e < E) {
        int r = rule_ids[e];
        int t = token_ids[e];
        atomicAdd(&pen[t], PENALTY_LAMBDA * gates[r]);
        firing[r] = 1.0f;   // racy same-value store: benign
    }
}

// ---------------------------------------------------------------------------
// Kernel 3: coverage loss via V_WMMA_F32_16X16X4_F32 reduction.
// One wave32, EXEC all-1s. A holds data (16x4 f32 = 2 VGPRs/lane), B = ones,
// so D[m,n] accumulates row-sums; chain C across R/64 chunks, then fold the
// 8 accumulator VGPRs + lane^16 to get the full sum (exact f32 for firing).
// ---------------------------------------------------------------------------
__global__ void dll_coverage_kernel(const float* __restrict__ gates,
                                    const float* __restrict__ firing,
                                    float* __restrict__ out_loss,
                                    int R) {
    int l  = threadIdx.x;            // 0..31, all lanes active (EXEC == ~0)
    int m  = l & 15;                 // matrix row
    int kb = (l >> 4) << 1;          // lanes 0-15: K=0,1; lanes 16-31: K=2,3

    v8f acc_gf = {};                 // accumulates sum(gates*firing)
    v8f acc_f  = {};                 // accumulates sum(firing)
    v2f ones;  ones[0] = 1.0f; ones[1] = 1.0f;   // all-ones B (layout-invariant)

    int chunks = R >> 6;             // 64 elements per 16x4 A tile
    for (int c = 0; c < chunks; ++c) {
        int i0 = (c << 6) + (m << 2) + kb;
        float f0 = firing[i0], f1 = firing[i0 + 1];
        v2f a_f, a_gf;
        a_f[0]  = f0;              a_f[1]  = f1;
        a_gf[0] = gates[i0] * f0;  a_gf[1] = gates[i0 + 1] * f1;
        acc_gf = __builtin_amdgcn_wmma_f32_16x16x4_f32(
            false, a_gf, false, ones, (short)0, acc_gf, false, false);
        acc_f  = __builtin_amdgcn_wmma_f32_16x16x4_f32(
            false, a_f,  false, ones, (short)0, acc_f,  false, false);
    }

    float s_gf = 0.0f, s_f = 0.0f;
#pragma unroll
    for (int i = 0; i < 8; ++i) { s_gf += acc_gf[i]; s_f += acc_f[i]; }
    // lane L holds D[rows 0-7, col L]; lane L+16 holds D[rows 8-15, col L]
    s_gf += __shfl_xor(s_gf, 16, 32);
    s_f  += __shfl_xor(s_f,  16, 32);

    if (l == 0) *out_loss = -s_gf / fmaxf(s_f, 1.0f);
}

// ---------------------------------------------------------------------------
// Kernel 4: streaming broadcast pass (the ~100 MB bandwidth-bound part).
// B128 NT loads of logits (read-once), RT loads of pen_1d (L2-hot, reused
// 64x), B128 NT stores for modified, scalar b32 NT stores for penalties
// (its base is offset by the scalar loss -> only 4B-aligned).
// ---------------------------------------------------------------------------
__global__ void dll_apply_kernel(const float* __restrict__ logits,
                                 const float* __restrict__ pen,
                                 float* __restrict__ modified,
                                 float* __restrict__ penalties,
                                 int V4) {
    int v4 = blockIdx.x * blockDim.x + threadIdx.x;   // 0..V4-1 (V/4)
    int b  = blockIdx.y;
    long base = (long)b * V4 + v4;                    // in float4 units

    f32x4 lg = __builtin_nontemporal_load((const f32x4*)logits + base);
    f32x4 p  = *((const f32x4*)pen + v4);             // regular (keep in L2)
    f32x4 md = lg - p;

    __builtin_nontemporal_store(md, (f32x4*)modified + base);

    float* pd = penalties + (base << 2);
    __builtin_nontemporal_store(p[0], pd + 0);
    __builtin_nontemporal_store(p[1], pd + 1);
    __builtin_nontemporal_store(p[2], pd + 2);
    __builtin_nontemporal_store(p[3], pd + 3);
}

// ---------------------------------------------------------------------------
extern "C" void kernel_launch(void* const* d_in, const int* in_sizes, int n_in,
                              void* d_out, int out_size, void* d_ws, size_t ws_size,
                              hipStream_t stream) {
    const float* logits      = (const float*)d_in[0];
    const float* gate_logits = (const float*)d_in[1];
    const int*   rule_ids    = (const int*)d_in[2];
    const int*   token_ids   = (const int*)d_in[3];

    const int BV = in_sizes[0];      // B * V = 8,192,000
    const int R  = in_sizes[1];      // 512
    const int E  = in_sizes[2];      // 1,000,000
    const int V  = 128000;
    const int B  = BV / V;

    float* out       = (float*)d_out;
    float* modified  = out;               // [B*V]
    float* loss      = out + BV;          // [1]
    float* penalties = out + BV + 1;      // [B*V], 4B-aligned only

    // workspace: gates[R] | firing[R] | pen_1d[V]  (~504 KB)
    float* gates  = (float*)d_ws;
    float* firing = gates + R;
    float* pen    = firing + R;

    dll_init_kernel<<<(R + V + 255) / 256, 256, 0, stream>>>(
        gate_logits, gates, firing, pen, R, V);

    dll_scatter_kernel<<<(E + 255) / 256, 256, 0, stream>>>(
        rule_ids, token_ids, gates, firing, pen, E);

    dll_coverage_kernel<<<1, 32, 0, stream>>>(gates, firing, loss, R);

    const int V4 = V / 4;                        // 32000
    dim3 grid(V4 / 256, B);                      // 125 x 64 blocks
    dll_apply_kernel<<<grid, 256, 0, stream>>>(
        logits, pen, modified, penalties, V4);
}